// Model_35673998361057
// MI455X (gfx1250) — compile-verified
//
#include <hip/hip_runtime.h>
#include <math.h>

#define N_EVENTS    16
#define N_SAMPLES   32768
#define STEP        256
#define LONG_WINDOW 4096
#define LONG_COEFFS 2049
#define LONG_STEP   2048
#define LONG_FRAMES 16
#define B_SZ        8
#define BE          128

typedef __attribute__((ext_vector_type(2))) float v2f;
typedef __attribute__((ext_vector_type(8))) float v8f;
typedef __attribute__((ext_vector_type(4))) unsigned int u32x4;
typedef __attribute__((ext_vector_type(4))) int i32x4;
typedef __attribute__((ext_vector_type(8))) int i32x8;

#define PI_F      3.14159265358979323846f
#define TWO_PI_F  6.28318530717958647692f

// ---------------------------------------------------------------------------
// Prologue 1: energy = env*(2*noise-1), transposed (b, n, f) -> (b, f, n)
// ---------------------------------------------------------------------------
__global__ void __launch_bounds__(512)
k_energy_transpose(const float* __restrict__ env,
                   const float* __restrict__ noise,
                   float* __restrict__ energy_t) {
  __shared__ float tile[32][17];
  int b = blockIdx.x;
  const float* envb = env   + (size_t)b * LONG_WINDOW * LONG_FRAMES;
  const float* nzb  = noise + (size_t)b * LONG_WINDOW * LONG_FRAMES;
  float* outb = energy_t + (size_t)b * LONG_FRAMES * LONG_WINDOW;
  int t = threadIdx.x;
  for (int nb = 0; nb < LONG_WINDOW; nb += 32) {
    int nl = t >> 4, f = t & 15;
    float e = envb[(nb + nl) * 16 + f];
    float z = nzb[(nb + nl) * 16 + f];
    tile[nl][f] = e * (z * 2.f - 1.f);
    __syncthreads();
    int fo = t >> 5, no = t & 31;
    outb[fo * LONG_WINDOW + nb + no] = tile[no][fo];
    __syncthreads();
  }
}

// ---------------------------------------------------------------------------
// Prologue 2: per-batch max of |pair| magnitude
// ---------------------------------------------------------------------------
__global__ void __launch_bounds__(256)
k_mag_max(const float* __restrict__ tf, float* __restrict__ mx) {
  __shared__ float red[256];
  int b = blockIdx.x;
  const float* tfb = tf + (size_t)b * (2 * LONG_COEFFS) * LONG_FRAMES;
  float m = 0.f;
  for (int i = threadIdx.x; i < LONG_COEFFS * LONG_FRAMES; i += 256) {
    int c = i >> 4, f = i & 15;
    float a  = tfb[(2 * c) * 16 + f];
    float bb = tfb[(2 * c + 1) * 16 + f];
    m = fmaxf(m, sqrtf(a * a + bb * bb));
  }
  red[threadIdx.x] = m;
  __syncthreads();
  for (int s = 128; s > 0; s >>= 1) {
    if (threadIdx.x < (unsigned)s)
      red[threadIdx.x] = fmaxf(red[threadIdx.x], red[threadIdx.x + s]);
    __syncthreads();
  }
  if (threadIdx.x == 0) mx[b] = red[0];
}

// ---------------------------------------------------------------------------
// Prologue 3: tfc = mag_n * exp(i * pi * atan2(imag, mag_n)), (b,c,f)->(b,f,c)
// ---------------------------------------------------------------------------
__global__ void __launch_bounds__(512)
k_tfc(const float* __restrict__ tf, const float* __restrict__ mx,
      float* __restrict__ tre, float* __restrict__ tim) {
  __shared__ float tr[32][17];
  __shared__ float ti[32][17];
  int b = blockIdx.x;
  const float* tfb = tf + (size_t)b * (2 * LONG_COEFFS) * LONG_FRAMES;
  float inv = 1.f / (mx[b] + 1e-8f);
  float* treb = tre + (size_t)b * LONG_FRAMES * LONG_COEFFS;
  float* timb = tim + (size_t)b * LONG_FRAMES * LONG_COEFFS;
  int t = threadIdx.x;
  for (int cb = 0; cb < LONG_COEFFS; cb += 32) {
    int cl = t >> 4, f = t & 15;
    int c = cb + cl;
    if (c < LONG_COEFFS) {
      float a  = tfb[(2 * c) * 16 + f];
      float bb = tfb[(2 * c + 1) * 16 + f];
      float magn = sqrtf(a * a + bb * bb) * inv;
      float im = tfb[(LONG_COEFFS + c) * 16 + f];
      float ph = atan2f(im, magn) * PI_F;
      float s, co; __sincosf(ph, &s, &co);
      tr[cl][f] = magn * co;
      ti[cl][f] = magn * s;
    }
    __syncthreads();
    int fo = t >> 5, co2 = t & 31;
    int c2 = cb + co2;
    if (c2 < LONG_COEFFS) {
      treb[fo * LONG_COEFFS + c2] = tr[co2][fo];
      timb[fo * LONG_COEFFS + c2] = ti[co2][fo];
    }
    __syncthreads();
  }
}

// ---------------------------------------------------------------------------
// TDM: 1D tile (nelem f32, contiguous) global -> LDS. D# per CDNA5 ISA ch.8.
// Issued by one wave; completion via TENSORcnt.
// ---------------------------------------------------------------------------
__device__ __forceinline__ unsigned int lds_byte_off(const void* p) {
  // generic LDS address: low 32 bits are the LDS byte address (ISA 10.2)
  return (unsigned int)(uintptr_t)p;
}

__device__ __forceinline__ void tdm_load_1d(const float* gsrc,
                                            unsigned int lds_byte,
                                            unsigned int nelem) {
  unsigned long long ga = (unsigned long long)(uintptr_t)gsrc;
  u32x4 g0;
  g0[0] = 1u;                                            // count=1, user D#
  g0[1] = lds_byte;                                      // lds_addr
  g0[2] = (unsigned int)(ga & 0xFFFFFFFFu);              // global_addr[31:0]
  g0[3] = (unsigned int)((ga >> 32) & 0x01FFFFFFu)       // global_addr[56:32]
        | (2u << 30);                                    // type=2 (image)
  i32x8 g1;
  g1[0] = (int)(2u << 16);                               // data_size=4B, mask=0
  g1[1] = (int)((nelem & 0xFFFFu) << 16);                // tensor_dim0[15:0]
  g1[2] = (int)(((nelem >> 16) & 0xFFFFu) | (1u << 16)); // dim0 hi | tensor_dim1=1
  g1[3] = (int)((nelem & 0xFFFFu) << 16);                // tile_dim0
  g1[4] = (int)1;                                        // tile_dim1=1
  g1[5] = (int)nelem;                                    // tensor_dim0_stride lo
  g1[6] = 0;
  g1[7] = 0;
  i32x4 z4 = {0, 0, 0, 0};
#if defined(__clang_major__) && (__clang_major__ >= 23)
  i32x8 z8 = {0, 0, 0, 0, 0, 0, 0, 0};
  __builtin_amdgcn_tensor_load_to_lds(g0, g1, z4, z4, z8, 0);
#else
  __builtin_amdgcn_tensor_load_to_lds(g0, g1, z4, z4, 0);
#endif
}

// ---------------------------------------------------------------------------
// One radix-16 stage of a 4096-pt FFT as WMMA 16x16 complex matmuls.
//   STAGE 0: B[m][col] = src[m*256 + t*16 + c], tw = w4096^(j*(t*16+c)),
//            store dst[j*256 + t*16 + c]
//   STAGE 1: B[m][c]   = src[t*256 + m*16 + c], tw = w256^(j*c),
//            store dst[t*256 + j*16 + c]
//   STAGE 2: B[m][c]   = src[t*256 + c*16 + m], no tw,
//            store dst[j*256 + c*16 + t]
// aimP = Im(F16) for this direction, aimN = -aimP (f32 WMMA has no A-neg).
// ---------------------------------------------------------------------------
template <int STAGE, bool INV>
__device__ __forceinline__ void stage_pass(const float* __restrict__ sre,
                                           const float* __restrict__ sim,
                                           float* __restrict__ dre,
                                           float* __restrict__ dim,
                                           const v2f* are, const v2f* aimP,
                                           const v2f* aimN) {
  int lane = threadIdx.x & 31;
  int t    = threadIdx.x >> 5;   // tile 0..15 (one per wave)
  int c    = lane & 15;
  int half = lane >> 4;
  v8f cre = {};
  v8f cim = {};
#pragma unroll
  for (int kk = 0; kk < 4; ++kk) {
    int m0 = 4 * kk + 2 * half;           // B layout: v0=row m0, v1=row m0+1
    int i0, d;
    if (STAGE == 0)      { i0 = m0 * 256 + t * 16 + c; d = 256; }
    else if (STAGE == 1) { i0 = t * 256 + m0 * 16 + c; d = 16;  }
    else                 { i0 = t * 256 + c * 16 + m0; d = 1;   }
    v2f br, bi;
    br.x = sre[i0]; br.y = sre[i0 + d];
    bi.x = sim[i0]; bi.y = sim[i0 + d];
    cre = __builtin_amdgcn_wmma_f32_16x16x4_f32(false, are[kk],  false, br, (short)0, cre, false, false);
    cre = __builtin_amdgcn_wmma_f32_16x16x4_f32(false, aimN[kk], false, bi, (short)0, cre, false, false);
    cim = __builtin_amdgcn_wmma_f32_16x16x4_f32(false, are[kk],  false, bi, (short)0, cim, false, false);
    cim = __builtin_amdgcn_wmma_f32_16x16x4_f32(false, aimP[kk], false, br, (short)0, cim, false, false);
  }
  const float dsign = INV ? 1.f : -1.f;
#pragma unroll
  for (int v = 0; v < 8; ++v) {
    int j = v + half * 8;                 // D layout: lanes16-31 hold M=v+8
    float xr = cre[v], xi = cim[v];
    if (STAGE == 0) {
      float th = (float)(j * (t * 16 + c)) * (TWO_PI_F / 4096.f);
      float s, co; __sincosf(th, &s, &co); s *= dsign;
      dre[j * 256 + t * 16 + c] = xr * co - xi * s;
      dim[j * 256 + t * 16 + c] = xr * s + xi * co;
    } else if (STAGE == 1) {
      float th = (float)(j * c) * (TWO_PI_F / 256.f);
      float s, co; __sincosf(th, &s, &co); s *= dsign;
      dre[t * 256 + j * 16 + c] = xr * co - xi * s;
      dim[t * 256 + j * 16 + c] = xr * s + xi * co;
    } else {
      dre[j * 256 + c * 16 + t] = xr;
      dim[j * 256 + c * 16 + t] = xi;
    }
  }
}

// ---------------------------------------------------------------------------
// Main scan: one workgroup per batch row; 16 sequential frames of
// (prev+energy) -> fwd FFT -> *tfc (Hermitian) -> inv FFT -> overlap-add.
// Energy tiles are double-buffered in LDS and prefetched by the TDM one
// frame ahead (wave 0 issues, TENSORcnt + barrier publishes).
// ---------------------------------------------------------------------------
__global__ void __launch_bounds__(512)
k_fft_scan(const float* __restrict__ energy_t,
           const float* __restrict__ tre,
           const float* __restrict__ tim,
           float* __restrict__ atoms) {
  extern __shared__ float smem[];
  float* Are = smem;             // 4096
  float* Aim = Are + 4096;       // 4096
  float* Bre = Aim + 4096;       // 4096
  float* Bim = Bre + 4096;       // 4096
  float* P   = Bim + 4096;       // 4096 (previous frame)
  float* Cy  = P + 4096;         // 2048 (overlap-add carry)
  float* Est = Cy + 2048;        // 2 x 4096 TDM staging (double buffer)

  int b    = blockIdx.x;
  int tid  = threadIdx.x;
  int lane = tid & 31;
  int row  = lane & 15;
  int half = lane >> 4;
  bool w0  = (tid >> 5) == 0;

  // DFT-16 matrix fragments in WMMA 16x4 A layout (lane<16: K=k0,k0+1; else +2)
  v2f fre[4], fimF[4], fimI[4];
#pragma unroll
  for (int kk = 0; kk < 4; ++kk) {
    int k0 = 4 * kk + 2 * half;
#pragma unroll
    for (int p = 0; p < 2; ++p) {
      int k = k0 + p;
      float th = (float)((row * k) & 15) * (TWO_PI_F / 16.f);
      float s, co; __sincosf(th, &s, &co);
      fre[kk][p]  = co;
      fimF[kk][p] = -s;          // forward: e^{-i th}
      fimI[kk][p] = s;           // inverse: e^{+i th}
    }
  }

  const float* eb  = energy_t + (size_t)b * LONG_FRAMES * LONG_WINDOW;
  const float* trb = tre + (size_t)b * LONG_FRAMES * LONG_COEFFS;
  const float* tib = tim + (size_t)b * LONG_FRAMES * LONG_COEFFS;
  float* ab = atoms + (size_t)b * N_SAMPLES;

  // Kick off TDM for frame 0's energy tile while we zero-init state.
  if (w0) tdm_load_1d(eb, lds_byte_off(Est), LONG_WINDOW);

  for (int i = tid; i < 4096; i += 512) P[i] = 0.f;
  for (int i = tid; i < 2048; i += 512) Cy[i] = 0.f;
  __syncthreads();

  for (int f = 0; f < LONG_FRAMES; ++f) {
    // Wait for this frame's staged energy tile, publish to all waves.
    if (w0) __builtin_amdgcn_s_wait_tensorcnt(0);
    __syncthreads();
    const float* Ecur = Est + (f & 1) * LONG_WINDOW;
    for (int i = tid; i < 4096; i += 512) {
      Are[i] = P[i] + Ecur[i];
      Aim[i] = 0.f;
    }
    // Prefetch next frame's tfc row (fire-and-forget, no counter).
    if (f + 1 < LONG_FRAMES && tid < 65) {
      __builtin_prefetch(trb + (f + 1) * LONG_COEFFS + tid * 32, 0, 0);
      __builtin_prefetch(tib + (f + 1) * LONG_COEFFS + tid * 32, 0, 0);
    }
    __syncthreads();
    // Overlap next frame's energy DMA with this frame's six WMMA stages.
    if (w0 && (f + 1 < LONG_FRAMES))
      tdm_load_1d(eb + (size_t)(f + 1) * LONG_WINDOW,
                  lds_byte_off(Est + ((f + 1) & 1) * LONG_WINDOW), LONG_WINDOW);

    stage_pass<0, false>(Are, Aim, Bre, Bim, fre, fimF, fimI); __syncthreads();
    stage_pass<1, false>(Bre, Bim, Are, Aim, fre, fimF, fimI); __syncthreads();
    stage_pass<2, false>(Are, Aim, Bre, Bim, fre, fimF, fimI); __syncthreads();

    // spectrum in B; multiply by tfc and Hermitian-extend into A
    const float* trf = trb + f * LONG_COEFFS;
    const float* tif = tib + f * LONG_COEFFS;
    for (int k = tid; k < LONG_COEFFS; k += 512) {
      float xr = Bre[k], xi = Bim[k];
      float t0 = trf[k], t1 = tif[k];
      float sr = xr * t0 - xi * t1;
      float si = xr * t1 + xi * t0;
      Are[k] = sr; Aim[k] = si;
      if (k >= 1 && k < 2048) { Are[4096 - k] = sr; Aim[4096 - k] = -si; }
    }
    __syncthreads();
    stage_pass<0, true>(Are, Aim, Bre, Bim, fre, fimI, fimF); __syncthreads();
    stage_pass<1, true>(Bre, Bim, Are, Aim, fre, fimI, fimF); __syncthreads();
    stage_pass<2, true>(Are, Aim, Bre, Bim, fre, fimI, fimF); __syncthreads();

    // frame = Re(inv)/4096 (ortho fwd * ortho inv); overlap-add with hop 2048
    for (int i = tid; i < 2048; i += 512) {
      float lo = Bre[i]        * (1.f / 4096.f);
      float hi = Bre[i + 2048] * (1.f / 4096.f);
      ab[f * 2048 + i] = Cy[i] + lo;
      Cy[i] = hi;
      P[i] = lo;
      P[i + 2048] = hi;
    }
    __syncthreads();
  }
}

// ---------------------------------------------------------------------------
// Scatter-add of 16 events per output row, then copy first N_SAMPLES
// ---------------------------------------------------------------------------
__global__ void k_zero(float* __restrict__ p, int n) {
  int i = blockIdx.x * 256 + threadIdx.x;
  if (i < n) p[i] = 0.f;
}

__global__ void __launch_bounds__(256)
k_scatter(const float* __restrict__ atoms, const int* __restrict__ idx,
          float* __restrict__ acc) {
  int blk = blockIdx.x;            // bb*16 + e
  int bb  = blk >> 4;
  int t0  = idx[blk] * STEP;       // indices[bb][e] * 256
  const float* x = atoms + (size_t)blk * N_SAMPLES;
  float* o = acc + (size_t)bb * (2 * N_SAMPLES) + t0;
  for (int s = threadIdx.x; s < N_SAMPLES; s += 256)
    atomicAdd(&o[s], x[s]);
}

__global__ void k_copy(const float* __restrict__ acc, float* __restrict__ out) {
  int i = blockIdx.x * 256 + threadIdx.x;
  if (i < B_SZ * N_SAMPLES) {
    int bb = i / N_SAMPLES;
    int s  = i - bb * N_SAMPLES;
    out[i] = acc[(size_t)bb * (2 * N_SAMPLES) + s];
  }
}

// ---------------------------------------------------------------------------
extern "C" void kernel_launch(void* const* d_in, const int* in_sizes, int n_in,
                              void* d_out, int out_size, void* d_ws, size_t ws_size,
                              hipStream_t stream) {
  const float* env     = (const float*)d_in[0];
  const float* tf      = (const float*)d_in[1];
  const float* noise   = (const float*)d_in[2];
  const int*   indices = (const int*)d_in[3];
  float* out = (float*)d_out;

  float* ws = (float*)d_ws;   // needs ~86 MB
  size_t off = 0;
  float* energy_t = ws + off; off += (size_t)BE * LONG_FRAMES * LONG_WINDOW; // 8,388,608 f
  float* tre      = ws + off; off += (size_t)BE * LONG_FRAMES * LONG_COEFFS; // 4,196,352 f
  float* tim      = ws + off; off += (size_t)BE * LONG_FRAMES * LONG_COEFFS; // 4,196,352 f
  float* mx       = ws + off; off += 128;
  float* atoms    = ws + off; off += (size_t)BE * N_SAMPLES;                 // 4,194,304 f
  float* acc      = ws + off; off += (size_t)B_SZ * 2 * N_SAMPLES;           //   524,288 f

  k_energy_transpose<<<BE, 512, 0, stream>>>(env, noise, energy_t);
  k_mag_max<<<BE, 256, 0, stream>>>(tf, mx);
  k_tfc<<<BE, 512, 0, stream>>>(tf, mx, tre, tim);

  // 4 FFT buffers + prev + carry + 2x energy staging
  size_t lds = (size_t)(4 * 4096 + 4096 + 2048 + 2 * 4096) * sizeof(float); // 122880 B
  k_fft_scan<<<BE, 512, lds, stream>>>(energy_t, tre, tim, atoms);

  int naz = B_SZ * 2 * N_SAMPLES;
  k_zero<<<(naz + 255) / 256, 256, 0, stream>>>(acc, naz);
  k_scatter<<<BE, 256, 0, stream>>>(atoms, indices, acc);
  int no = B_SZ * N_SAMPLES;
  k_copy<<<(no + 255) / 256, 256, 0, stream>>>(acc, out);
}